// Aggregator_48971217109579
// MI455X (gfx1250) — compile-verified
//
#include <hip/hip_runtime.h>

#define CH 128          // channels per node (float32)
#define CH4 (CH / 4)    // float4s per row == 32 == wave32 lanes
#define MAX_REL 24

typedef int v4i __attribute__((vector_size(16)));
typedef __attribute__((address_space(1))) v4i* gv4i_ptr;   // global
typedef __attribute__((address_space(3))) v4i* lv4i_ptr;   // LDS

// ---------------------------------------------------------------------------
// Zero-init the output accumulator with 128-bit stores.
// ---------------------------------------------------------------------------
__global__ __launch_bounds__(256) void zero_out_kernel(float4* __restrict__ out, int n4) {
    int i = blockIdx.x * blockDim.x + threadIdx.x;
    if (i < n4) out[i] = make_float4(0.f, 0.f, 0.f, 0.f);
}

// ---------------------------------------------------------------------------
// RGCN-style edge aggregation:
//   out[head[e], :] += all_emb[tail[e], :] * weight[etype[e], :]
//
// One wave32 per edge per iteration: lane L owns channels [4L, 4L+4).
//  - relation table staged in LDS via gfx1250 async-to-LDS (ASYNCcnt) when
//    the builtin exists; plain load+ds_store fallback otherwise
//  - edge metadata loads use a readfirstlane'd edge id -> uniform address ->
//    SMEM s_load path (KMcnt), keeping the vector pipe free for the gather
//  - software-pipelined: next edge's (h,t,r) is fetched one iteration early
//    and its 512B embedding row is warmed with global_prefetch_b8
//  - gather: global_load_b128 (all_emb is 25.6 MB, resident in 192 MB L2)
//  - scatter: 4x global_atomic_add_f32, no-return variant (STOREcnt, L2)
// ---------------------------------------------------------------------------
__global__ __launch_bounds__(256) void rgcn_agg_kernel(
    const float* __restrict__ all_emb,
    const int*   __restrict__ head,
    const int*   __restrict__ tail,
    const int*   __restrict__ etype,
    const float* __restrict__ weight,
    float*       __restrict__ out,
    int n_edges, int n_rel)
{
    __shared__ float4 w_lds[MAX_REL * CH4];   // 24 * 32 * 16B = 12 KB

    const int lane          = threadIdx.x & 31;
    const int wave_in_blk   = threadIdx.x >> 5;
    const int waves_per_blk = blockDim.x >> 5;

    // ---- Stage relation weights into LDS ---------------------------------
    const float4* wsrc = reinterpret_cast<const float4*>(weight);
#if __has_builtin(__builtin_amdgcn_global_load_async_to_lds_b128)
    for (int i = threadIdx.x; i < n_rel * CH4; i += blockDim.x) {
        v4i* gp = (v4i*)(wsrc + i);    // C-style: reinterpret + drop const
        v4i* lp = (v4i*)(w_lds + i);
        __builtin_amdgcn_global_load_async_to_lds_b128(
            (gv4i_ptr)gp, (lv4i_ptr)lp, 0, 0);
    }
  #if __has_builtin(__builtin_amdgcn_s_wait_asynccnt)
    __builtin_amdgcn_s_wait_asynccnt(0);
  #else
    asm volatile("s_wait_asynccnt 0x0" ::: "memory");
  #endif
#else
    for (int i = threadIdx.x; i < n_rel * CH4; i += blockDim.x)
        w_lds[i] = wsrc[i];
#endif
    __syncthreads();

    const int wave   = blockIdx.x * waves_per_blk + wave_in_blk;
    const int nwaves = gridDim.x * waves_per_blk;

    // ---- Software-pipelined edge loop ------------------------------------
    int h = 0, t = 0, r = 0;
    if (wave < n_edges) {
        const int eu = __builtin_amdgcn_readfirstlane(wave);  // uniform addr
        h = head[eu];
        t = tail[eu];
        r = etype[eu];
    }

    for (int e = wave; e < n_edges; e += nwaves) {
        // Fetch next edge's metadata early and warm its embedding row while
        // the current gather + atomics are in flight.
        int hn = 0, tn = 0, rn = 0;
        const int en = e + nwaves;
        if (en < n_edges) {
            const int eun = __builtin_amdgcn_readfirstlane(en);
            hn = head[eun];
            tn = tail[eun];
            rn = etype[eun];
            // 32 lanes x 16B = the whole 512B row of the next gather
            __builtin_prefetch(all_emb + (size_t)tn * CH + (lane << 2), 0, 3);
        }

        // Gather tail row: 32 lanes x float4 = 512B contiguous (b128).
        const float4 x = reinterpret_cast<const float4*>(
                             all_emb + (size_t)t * CH)[lane];
        // Relation row from LDS (ds_load_b128, conflict-free).
        const float4 w = w_lds[r * CH4 + lane];

        // Scatter-add into the head row: hardware fp32 atomics in L2.
        float* dst = out + (size_t)h * CH + (lane << 2);
        unsafeAtomicAdd(dst + 0, x.x * w.x);
        unsafeAtomicAdd(dst + 1, x.y * w.y);
        unsafeAtomicAdd(dst + 2, x.z * w.z);
        unsafeAtomicAdd(dst + 3, x.w * w.w);

        h = hn; t = tn; r = rn;
    }
}

// ---------------------------------------------------------------------------
// Harness entry point.
//   d_in[0] = all_emb   [N_NODES * 128] f32
//   d_in[1] = edge_index[2 * E]         int (row-major: head row, tail row)
//   d_in[2] = edge_type [E]             int
//   d_in[3] = weight    [N_REL * 128]   f32
//   d_out   = res_emb   [N_NODES * 128] f32
// ---------------------------------------------------------------------------
extern "C" void kernel_launch(void* const* d_in, const int* in_sizes, int n_in,
                              void* d_out, int out_size, void* d_ws, size_t ws_size,
                              hipStream_t stream)
{
    const float* all_emb = (const float*)d_in[0];
    const int*   eidx    = (const int*)d_in[1];
    const int*   etype   = (const int*)d_in[2];
    const float* weight  = (const float*)d_in[3];
    float*       out     = (float*)d_out;

    const int n_edges = in_sizes[1] / 2;
    int       n_rel   = in_sizes[3] / CH;
    if (n_rel > MAX_REL) n_rel = MAX_REL;

    const int* head = eidx;            // edge_index[0, :]
    const int* tail = eidx + n_edges;  // edge_index[1, :]

    // 1) zero the accumulator (output is poisoned before timing)
    const int n4 = out_size / 4;
    zero_out_kernel<<<(n4 + 255) / 256, 256, 0, stream>>>((float4*)out, n4);

    // 2) aggregate. 4096 blocks x 8 waves = 32768 waves, ~18 edges each;
    //    12 KB LDS staging amortized, high occupancy across ~1024 SIMD32s.
    int blocks = (n_edges + 7) / 8;    // one wave per edge upper bound
    if (blocks > 4096) blocks = 4096;
    if (blocks < 1) blocks = 1;
    rgcn_agg_kernel<<<blocks, 256, 0, stream>>>(
        all_emb, head, tail, etype, weight, out, n_edges, n_rel);
}